// GATLayer_88579405512952
// MI455X (gfx1250) — compile-verified
//
#include <hip/hip_runtime.h>
#include <hip/hip_bf16.h>
#include <cstddef>

#define N_ROWS   8192
#define IN_DIM   256
#define OUT_DIM  64
#define ALPHA    0.2f
#define CHUNK    64
#define WSTRIDE  68   // LDS row stride (floats): conflict-free b64 A-reads
#define RPB      32   // rows per block (2 x 16-row WMMA tiles per wave)

typedef __attribute__((ext_vector_type(2))) float v2f;
typedef __attribute__((ext_vector_type(8))) float v8f;

__device__ __forceinline__ float leaky(float x) {
    return x > 0.0f ? x : ALPHA * x;
}

// ---------------------------------------------------------------------------
// Kernel 1: Wh = h @ W ; s = Wh @ a[:64] ; t = Wh @ a[64:]
// One 64-thread block per row; thread c owns output column c.
// ---------------------------------------------------------------------------
__global__ __launch_bounds__(64) void gat_wh_kernel(
    const float* __restrict__ h, const float* __restrict__ W,
    const float* __restrict__ a, float* __restrict__ wh,
    float* __restrict__ s, float* __restrict__ t)
{
    __shared__ float red1[64];
    __shared__ float red2[64];
    const int i = blockIdx.x;
    const int c = threadIdx.x;

    const float* hp = h + (size_t)i * IN_DIM;
    float acc = 0.0f;
    #pragma unroll 8
    for (int k = 0; k < IN_DIM; ++k)
        acc = fmaf(hp[k], W[k * OUT_DIM + c], acc);   // hp[k] uniform -> s_load

    wh[(size_t)i * OUT_DIM + c] = acc;
    red1[c] = acc * a[c];
    red2[c] = acc * a[OUT_DIM + c];
    __syncthreads();
    #pragma unroll
    for (int off = 32; off > 0; off >>= 1) {
        if (c < off) { red1[c] += red1[c + off]; red2[c] += red2[c + off]; }
        __syncthreads();
    }
    if (c == 0) { s[i] = red1[0]; t[i] = red2[0]; }
}

// ---------------------------------------------------------------------------
// Kernel 2: tmax = max_j t[j]   (softmax stabilizer; leaky is monotone)
// ---------------------------------------------------------------------------
__global__ __launch_bounds__(256) void gat_tmax_kernel(
    const float* __restrict__ t, float* __restrict__ tmaxp)
{
    __shared__ float red[256];
    const int tid = threadIdx.x;
    float m = -3.0e38f;
    for (int j = tid; j < N_ROWS; j += 256) m = fmaxf(m, t[j]);
    red[tid] = m;
    __syncthreads();
    #pragma unroll
    for (int off = 128; off > 0; off >>= 1) {
        if (tid < off) red[tid] = fmaxf(red[tid], red[tid + off]);
        __syncthreads();
    }
    if (tid == 0) tmaxp[0] = red[0];
}

// ---------------------------------------------------------------------------
// Kernel 3: out = elu( softmax_row(leaky(s_i + t_j)) @ Wh )
// 256 blocks x 128 threads (4 waves). Block -> 32 rows; wave w -> cols 16w..16w+15
// with TWO 16x16 f32 accumulators sharing each B fragment. Exp-weight tiles are
// DOUBLE-BUFFERED in LDS: chunk n+1's weights are generated while chunk n feeds
// v_wmma_f32_16x16x4_f32 -> one barrier per chunk, VALU/exp co-executes with WMMA.
// ---------------------------------------------------------------------------
__global__ __launch_bounds__(128) void gat_main_kernel(
    const float* __restrict__ wh, const float* __restrict__ s,
    const float* __restrict__ t,  const float* __restrict__ tmaxp,
    float* __restrict__ out)
{
    __shared__ float wlds[2][RPB][WSTRIDE];
    __shared__ float zlds[RPB];

    const int tid  = threadIdx.x;
    const int wave = tid >> 5;          // 0..3 : output column block
    const int lane = tid & 31;          // wave32 lane
    const int half = lane >> 4;         // A/B fragment half
    const int lrow = lane & 15;
    const int rowbase = blockIdx.x * RPB;
    const int c0 = wave * 16;

    if (tid < RPB) zlds[tid] = 0.0f;

    // weight-generation role: thread covers row r, j-slice jg*16.. of each chunk
    const int r  = tid & (RPB - 1);     // 0..31
    const int jg = tid >> 5;            // 0..3 (16 j's each)
    const float tmax = tmaxp[0];
    const float sr = s[rowbase + r];
    const float mr = leaky(sr + tmax);  // row max of e (leaky monotone)
    float z0 = 0.0f, z1 = 0.0f, z2 = 0.0f, z3 = 0.0f;  // break the sum chain

    // generate one thread's 16 weights of chunk at jbase into buffer wb
    auto gen_w = [&](int jbase, float (*wb)[WSTRIDE]) {
        const float4* tp = reinterpret_cast<const float4*>(t + jbase + jg * 16);
        #pragma unroll
        for (int q4 = 0; q4 < 4; ++q4) {
            const float4 tv = tp[q4];
            const float w0 = __expf(leaky(sr + tv.x) - mr);   // v_exp_f32, w <= 1
            const float w1 = __expf(leaky(sr + tv.y) - mr);
            const float w2 = __expf(leaky(sr + tv.z) - mr);
            const float w3 = __expf(leaky(sr + tv.w) - mr);
            const int jl = jg * 16 + q4 * 4;
            wb[r][jl + 0] = w0;
            wb[r][jl + 1] = w1;
            wb[r][jl + 2] = w2;
            wb[r][jl + 3] = w3;
            z0 += w0; z1 += w1; z2 += w2; z3 += w3;
        }
    };

    v8f acc0 = {};                      // rows rowbase+0..15
    v8f acc1 = {};                      // rows rowbase+16..31

    gen_w(0, wlds[0]);                  // prologue: chunk 0 weights
    __syncthreads();

    int buf = 0;
    for (int j0 = 0; j0 < N_ROWS; j0 += CHUNK) {
        // produce next chunk's weights into the other buffer (overlaps with WMMA)
        if (j0 + CHUNK < N_ROWS) gen_w(j0 + CHUNK, wlds[buf ^ 1]);

        float (*wb)[WSTRIDE] = wlds[buf];
        #pragma unroll
        for (int jk = 0; jk < CHUNK; jk += 4) {
            // A (16x4): lanes 0-15 hold {K0,K1}, lanes 16-31 hold {K2,K3}
            v2f A0, A1, B;
            A0.x = wb[lrow][jk + 2 * half];
            A0.y = wb[lrow][jk + 2 * half + 1];
            A1.x = wb[16 + lrow][jk + 2 * half];
            A1.y = wb[16 + lrow][jk + 2 * half + 1];
            // B (4x16): VGPR0 = K0(l0-15)/K2(l16-31), VGPR1 = K1/K3
            const float* bp = wh + (size_t)(j0 + jk + 2 * half) * OUT_DIM + c0 + lrow;
            B.x = bp[0];
            B.y = bp[OUT_DIM];
            acc0 = __builtin_amdgcn_wmma_f32_16x16x4_f32(
                false, A0, false, B, (short)0, acc0, false, false);
            acc1 = __builtin_amdgcn_wmma_f32_16x16x4_f32(
                false, A1, false, B, (short)0, acc1, false, false);
        }
        __syncthreads();                // reads of wlds[buf] done; writes of buf^1 visible
        buf ^= 1;
    }

    atomicAdd(&zlds[r], (z0 + z1) + (z2 + z3));   // ds_add_f32; 4 partials per row
    __syncthreads();

    // C layout: VGPR v -> row v (lanes 0-15) / row v+8 (lanes 16-31), col = lane%16
    #pragma unroll
    for (int v = 0; v < 8; ++v) {
        const int rl = v + 8 * half;
        const float x0 = acc0[v] / zlds[rl];
        const float y0 = x0 > 0.0f ? x0 : (__expf(x0) - 1.0f);   // ELU(alpha=1)
        out[(size_t)(rowbase + rl) * OUT_DIM + c0 + lrow] = y0;
        const float x1 = acc1[v] / zlds[16 + rl];
        const float y1 = x1 > 0.0f ? x1 : (__expf(x1) - 1.0f);
        out[(size_t)(rowbase + 16 + rl) * OUT_DIM + c0 + lrow] = y1;
    }
}

// ---------------------------------------------------------------------------
extern "C" void kernel_launch(void* const* d_in, const int* in_sizes, int n_in,
                              void* d_out, int out_size, void* d_ws, size_t ws_size,
                              hipStream_t stream) {
    const float* h = (const float*)d_in[0];
    // d_in[1] = adj : unused by the reference computation
    const float* W = (const float*)d_in[2];
    const float* a = (const float*)d_in[3];
    float* out = (float*)d_out;

    float* ws  = (float*)d_ws;
    float* s   = ws;                                   // 8192
    float* t   = ws + N_ROWS;                          // 8192
    float* wh  = ws + 2 * N_ROWS;                      // 8192*64
    float* tmx = ws + 2 * N_ROWS + N_ROWS * OUT_DIM;   // 1

    gat_wh_kernel<<<N_ROWS, 64, 0, stream>>>(h, W, a, wh, s, t);
    gat_tmax_kernel<<<1, 256, 0, stream>>>(t, tmx);
    gat_main_kernel<<<N_ROWS / RPB, 128, 0, stream>>>(wh, s, t, tmx, out);
}